// HierarchicalPooling_73787538145381
// MI455X (gfx1250) — compile-verified
//
#include <hip/hip_runtime.h>

typedef __attribute__((ext_vector_type(2))) float v2f;
typedef __attribute__((ext_vector_type(8))) float v8f;

#define DH 128
#define SORT_TILE 4096          // keys per LDS tile (32 KB of u64 — WGP has 320 KB)
#define SORT_THREADS 512

// ---------------------------------------------------------------------------
// GEMM: H = X @ W   (X: n x 128, W: 128 x 128, H: n x 128), fp32 WMMA 16x16x4
// One wave32 per 16x16 output tile. n must be a multiple of 16 (50000/40000 are).
// A 16x4 f32 layout: lanes 0-15 -> M=0..15, VGPR0=K0,VGPR1=K1; lanes 16-31 -> K2,K3.
// B 4x16 f32 layout: VGPR0: K=0(lanes0-15)/K=2(lanes16-31); VGPR1: K=1/K=3.
// C/D 16x16 f32: VGPR r: M=r (lanes0-15), M=r+8 (lanes16-31); N = lane&15.
// ---------------------------------------------------------------------------
__global__ void __launch_bounds__(32)
gemm_wmma_f32(const float* __restrict__ X, const float* __restrict__ W,
              float* __restrict__ H) {
  const int tile    = blockIdx.x;
  const int colTile = tile & 7;          // 128/16 = 8 column tiles
  const int rowTile = tile >> 3;
  const int lane    = threadIdx.x;
  const int m       = lane & 15;
  const int kb      = (lane >> 4) << 1;  // 0 or 2
  const float* __restrict__ xr = X + (size_t)(rowTile * 16 + m) * DH;
  const int wc = colTile * 16 + m;

  v8f acc = {};
#pragma unroll 8
  for (int k0 = 0; k0 < DH; k0 += 4) {
    v2f a, b;
    a.x = xr[k0 + kb];
    a.y = xr[k0 + kb + 1];
    b.x = W[(size_t)(k0 + kb) * DH + wc];
    b.y = W[(size_t)(k0 + kb + 1) * DH + wc];
    acc = __builtin_amdgcn_wmma_f32_16x16x4_f32(false, a, false, b,
                                                (short)0, acc, false, false);
  }
  const int rowHi = (lane >> 4) << 3;    // 0 or 8
  float* __restrict__ out = H + (size_t)(rowTile * 16 + rowHi) * DH + colTile * 16 + m;
#pragma unroll
  for (int r = 0; r < 8; ++r) out[(size_t)r * DH] = acc[r];
}

// ---------------------------------------------------------------------------
// Degree count: deg[dst[e]] += 1 for live edges (integer -> deterministic).
// ---------------------------------------------------------------------------
__global__ void deg_count(const int* __restrict__ dst, const int* __restrict__ mask,
                          unsigned* __restrict__ deg, int E) {
  int e = blockIdx.x * blockDim.x + threadIdx.x;
  if (e >= E) return;
  if (mask && mask[e] == 0) return;
  atomicAdd(&deg[dst[e]], 1u);
}

// ---------------------------------------------------------------------------
// Edge scatter: agg[dst] += h[src] * dinv[src]*dinv[dst].
// thread = (edge, 4 features): float4 gather + 4 float atomics.
// ---------------------------------------------------------------------------
__global__ void gcn_scatter(const float* __restrict__ H, const int* __restrict__ src,
                            const int* __restrict__ dst, const unsigned* __restrict__ deg,
                            const int* __restrict__ mask, float* __restrict__ agg, int E) {
  int idx = blockIdx.x * blockDim.x + threadIdx.x;
  int e = idx >> 5;            // 32 threads per edge
  int q = idx & 31;            // quad index (4 floats each)
  if (e >= E) return;
  if (mask && mask[e] == 0) return;
  int s = src[e], d = dst[e];
  float ds = (float)deg[s] + 1.0f;
  float dd = (float)deg[d] + 1.0f;
  float norm = rsqrtf(ds) * rsqrtf(dd);
  float4 hv = reinterpret_cast<const float4*>(H + (size_t)s * DH)[q];
  float* a = agg + (size_t)d * DH + q * 4;
  atomicAdd(a + 0, hv.x * norm);
  atomicAdd(a + 1, hv.y * norm);
  atomicAdd(a + 2, hv.z * norm);
  atomicAdd(a + 3, hv.w * norm);
}

// ---------------------------------------------------------------------------
// Finish conv: h = relu(agg + h/deg + b)  (in place on H), float4 per thread.
// ---------------------------------------------------------------------------
__global__ void gcn_finish(float* __restrict__ H, const float* __restrict__ agg,
                           const unsigned* __restrict__ deg, const float* __restrict__ bias,
                           int n) {
  int idx = blockIdx.x * blockDim.x + threadIdx.x;   // over n*32 quads
  int i = idx >> 5;
  int q = idx & 31;
  if (i >= n) return;
  float dg = (float)deg[i] + 1.0f;
  float dinv = rsqrtf(dg);
  float sc = dinv * dinv;
  float4 hv = reinterpret_cast<float4*>(H + (size_t)i * DH)[q];
  float4 av = reinterpret_cast<const float4*>(agg + (size_t)i * DH)[q];
  float4 bv = reinterpret_cast<const float4*>(bias)[q];
  float4 r;
  r.x = av.x + hv.x * sc + bv.x; r.x = r.x > 0.0f ? r.x : 0.0f;
  r.y = av.y + hv.y * sc + bv.y; r.y = r.y > 0.0f ? r.y : 0.0f;
  r.z = av.z + hv.z * sc + bv.z; r.z = r.z > 0.0f ? r.z : 0.0f;
  r.w = av.w + hv.w * sc + bv.w; r.w = r.w > 0.0f ? r.w : 0.0f;
  reinterpret_cast<float4*>(H + (size_t)i * DH)[q] = r;
}

// ---------------------------------------------------------------------------
// Node score: score[i] = tanh( dot(h[i], attn) / ||attn|| ). 8 waves per block,
// one wave per node.
// ---------------------------------------------------------------------------
__global__ void __launch_bounds__(256)
node_score(const float* __restrict__ H, const float* __restrict__ attn,
           float* __restrict__ score, int n) {
  int node = blockIdx.x * 8 + (threadIdx.x >> 5);
  int lane = threadIdx.x & 31;
  if (node >= n) return;
  float dot = 0.0f, nrm = 0.0f;
  for (int f = lane; f < DH; f += 32) {
    float a = attn[f];
    nrm += a * a;
    dot += H[(size_t)node * DH + f] * a;
  }
  for (int off = 16; off; off >>= 1) {
    dot += __shfl_xor(dot, off, 32);
    nrm += __shfl_xor(nrm, off, 32);
  }
  if (lane == 0) score[node] = tanhf(dot * rsqrtf(nrm));
}

// ---------------------------------------------------------------------------
// TopK via 64-bit-key bitonic sort (descending, ties -> smallest index first,
// matching jax.lax.top_k). LDS-fused for spans <= SORT_TILE/2.
// ---------------------------------------------------------------------------
__device__ __forceinline__ unsigned long long packKey(float s, unsigned idx) {
  unsigned u = __float_as_uint(s);
  u = (u & 0x80000000u) ? ~u : (u | 0x80000000u);   // monotone float -> uint
  return ((unsigned long long)u << 32) | (0xFFFFFFFFu - idx);
}

__global__ void build_keys(const float* __restrict__ score,
                           unsigned long long* __restrict__ keys, int n, int npad) {
  int i = blockIdx.x * blockDim.x + threadIdx.x;
  if (i >= npad) return;
  keys[i] = (i < n) ? packKey(score[i], (unsigned)i) : 0ULL;
}

__device__ __forceinline__ int ceIndex(int t, int j) {
  // insert a 0 bit at the bit position of j: lower partner of compare-exchange t
  return ((t & ~(j - 1)) << 1) | (t & (j - 1));
}

// global compare-exchange step (span j >= SORT_TILE), grid = npad/2 threads
__global__ void bitonic_step(unsigned long long* __restrict__ keys, int j, int k, int half) {
  int t = blockIdx.x * blockDim.x + threadIdx.x;
  if (t >= half) return;
  int i = ceIndex(t, j);
  unsigned long long a = keys[i], b = keys[i ^ j];
  bool doSwap = ((i & k) == 0) ? (a < b) : (a > b);   // descending overall
  if (doSwap) { keys[i] = b; keys[i ^ j] = a; }
}

// full local sort of each SORT_TILE tile: all stages k = 2..SORT_TILE in LDS
__global__ void __launch_bounds__(SORT_THREADS)
bitonic_local_full(unsigned long long* __restrict__ keys) {
  __shared__ unsigned long long lds[SORT_TILE];
  const int base = blockIdx.x * SORT_TILE;
  for (int t = threadIdx.x; t < SORT_TILE; t += SORT_THREADS)
    lds[t] = keys[base + t];
  __syncthreads();
  for (int k = 2; k <= SORT_TILE; k <<= 1) {
    for (int j = k >> 1; j > 0; j >>= 1) {
      for (int t = threadIdx.x; t < SORT_TILE / 2; t += SORT_THREADS) {
        int i = ceIndex(t, j);
        int gi = base + i;
        unsigned long long a = lds[i], b = lds[i ^ j];
        bool doSwap = ((gi & k) == 0) ? (a < b) : (a > b);
        if (doSwap) { lds[i] = b; lds[i ^ j] = a; }
      }
      __syncthreads();
    }
  }
  for (int t = threadIdx.x; t < SORT_TILE; t += SORT_THREADS)
    keys[base + t] = lds[t];
}

// merge tail of stage k: all spans j = SORT_TILE/2 .. 1 in LDS
__global__ void __launch_bounds__(SORT_THREADS)
bitonic_local_merge(unsigned long long* __restrict__ keys, int k) {
  __shared__ unsigned long long lds[SORT_TILE];
  const int base = blockIdx.x * SORT_TILE;
  for (int t = threadIdx.x; t < SORT_TILE; t += SORT_THREADS)
    lds[t] = keys[base + t];
  __syncthreads();
  for (int j = SORT_TILE >> 1; j > 0; j >>= 1) {
    for (int t = threadIdx.x; t < SORT_TILE / 2; t += SORT_THREADS) {
      int i = ceIndex(t, j);
      int gi = base + i;
      unsigned long long a = lds[i], b = lds[i ^ j];
      bool doSwap = ((gi & k) == 0) ? (a < b) : (a > b);
      if (doSwap) { lds[i] = b; lds[i ^ j] = a; }
    }
    __syncthreads();
  }
  for (int t = threadIdx.x; t < SORT_TILE; t += SORT_THREADS)
    keys[base + t] = lds[t];
}

// ---------------------------------------------------------------------------
// Pool gather: xnew[i] = h[perm[i]] * score[perm[i]]; also records newid.
// thread = (kept node, 4 features)
// ---------------------------------------------------------------------------
__global__ void pool_gather(const float* __restrict__ H, const float* __restrict__ score,
                            const unsigned long long* __restrict__ keys,
                            float* __restrict__ xnew, int* __restrict__ newid, int kkeep) {
  int idx = blockIdx.x * blockDim.x + threadIdx.x;
  int i = idx >> 5;
  int q = idx & 31;
  if (i >= kkeep) return;
  unsigned srcIdx = 0xFFFFFFFFu - (unsigned)(keys[i] & 0xFFFFFFFFull);
  float val = score[srcIdx];
  float4 hv = reinterpret_cast<const float4*>(H + (size_t)srcIdx * DH)[q];
  float4 r; r.x = hv.x * val; r.y = hv.y * val; r.z = hv.z * val; r.w = hv.w * val;
  reinterpret_cast<float4*>(xnew + (size_t)i * DH)[q] = r;
  if (q == 0 && newid) newid[srcIdx] = i;
}

// ---------------------------------------------------------------------------
// Edge remap after pooling: keep edge if both endpoints kept & previously live.
// ---------------------------------------------------------------------------
__global__ void edge_remap(const int* __restrict__ src, const int* __restrict__ dst,
                           const int* __restrict__ maskIn, const int* __restrict__ newid,
                           int* __restrict__ src2, int* __restrict__ dst2,
                           int* __restrict__ mask2, int E) {
  int e = blockIdx.x * blockDim.x + threadIdx.x;
  if (e >= E) return;
  int s = src[e], d = dst[e];
  bool live = maskIn ? (maskIn[e] != 0) : true;
  int ns = newid[s], nd = newid[d];
  live = live && (ns >= 0) && (nd >= 0);
  src2[e]  = live ? ns : 0;
  dst2[e]  = live ? nd : 0;
  mask2[e] = live ? 1 : 0;
}

// ---------------------------------------------------------------------------
extern "C" void kernel_launch(void* const* d_in, const int* in_sizes, int n_in,
                              void* d_out, int out_size, void* d_ws, size_t ws_size,
                              hipStream_t stream) {
  (void)n_in; (void)out_size; (void)ws_size;
  const float* x     = (const float*)d_in[0];
  const int*   ei    = (const int*)  d_in[1];
  // d_in[2] = batch (unused; single graph)
  const float* W1    = (const float*)d_in[3];
  const float* b1    = (const float*)d_in[4];
  const float* attn1 = (const float*)d_in[5];
  const float* W2    = (const float*)d_in[6];
  const float* b2    = (const float*)d_in[7];
  const float* attn2 = (const float*)d_in[8];

  const int N  = in_sizes[0] / DH;            // 50000
  const int E  = in_sizes[1] / 2;             // 500000
  const int K1 = (int)((4LL * N  + 4) / 5);   // ceil(0.8N)  = 40000
  const int K2 = (int)((4LL * K1 + 4) / 5);   // ceil(0.8K1) = 32000
  const int* src1 = ei;
  const int* dst1 = ei + E;

  int npad = SORT_TILE; while (npad < N) npad <<= 1;  // 65536

  // workspace carve-out (256B aligned)
  char* ws = (char*)d_ws;
  size_t off = 0;
  auto carve = [&](size_t bytes) -> void* {
    void* p = ws + off;
    off += (bytes + 255) & ~(size_t)255;
    return p;
  };
  float*              bufA  = (float*)carve((size_t)N * DH * sizeof(float)); // h1 -> h2
  float*              bufB  = (float*)carve((size_t)N * DH * sizeof(float)); // agg1 -> x2 -> agg2
  unsigned*           deg   = (unsigned*)carve((size_t)N * sizeof(unsigned));
  float*              score = (float*)carve((size_t)N * sizeof(float));
  unsigned long long* keys  = (unsigned long long*)carve((size_t)npad * 8);
  int*                newid = (int*)carve((size_t)N * sizeof(int));
  int*                src2  = (int*)carve((size_t)E * sizeof(int));
  int*                dst2  = (int*)carve((size_t)E * sizeof(int));
  int*                mask2 = (int*)carve((size_t)E * sizeof(int));

  const int TB = 256;
  auto run_level = [&](const float* xin, const float* Wt, const float* bt,
                       const float* at, const int* srcE, const int* dstE,
                       const int* maskE, int n, int kkeep,
                       float* Hbuf, float* Abuf, float* poolOut, int* newidOut) {
    // h = xin @ Wt   (WMMA fp32 16x16x4)
    gemm_wmma_f32<<<(n / 16) * 8, 32, 0, stream>>>(xin, Wt, Hbuf);
    // degrees (integer counts; edge weights are exactly 0/1 throughout)
    hipMemsetAsync(deg, 0, (size_t)n * sizeof(unsigned), stream);
    deg_count<<<(E + TB - 1) / TB, TB, 0, stream>>>(dstE, maskE, deg, E);
    // edge aggregation
    hipMemsetAsync(Abuf, 0, (size_t)n * DH * sizeof(float), stream);
    int totalEQ = E * 32;
    gcn_scatter<<<(totalEQ + TB - 1) / TB, TB, 0, stream>>>(Hbuf, srcE, dstE, deg,
                                                            maskE, Abuf, E);
    // self loop + bias + relu (in place)
    gcn_finish<<<(n * 32 + TB - 1) / TB, TB, 0, stream>>>(Hbuf, Abuf, deg, bt, n);
    // scores
    node_score<<<(n + 7) / 8, 256, 0, stream>>>(Hbuf, at, score, n);
    // top-k: descending bitonic sort of packed (score, ~idx) keys
    build_keys<<<(npad + TB - 1) / TB, TB, 0, stream>>>(score, keys, n, npad);
    bitonic_local_full<<<npad / SORT_TILE, SORT_THREADS, 0, stream>>>(keys);
    for (int k = SORT_TILE << 1; k <= npad; k <<= 1) {
      for (int j = k >> 1; j >= SORT_TILE; j >>= 1)
        bitonic_step<<<(npad / 2 + TB - 1) / TB, TB, 0, stream>>>(keys, j, k, npad / 2);
      bitonic_local_merge<<<npad / SORT_TILE, SORT_THREADS, 0, stream>>>(keys, k);
    }
    if (newidOut) hipMemsetAsync(newidOut, 0xFF, (size_t)n * sizeof(int), stream);
    pool_gather<<<(kkeep * 32 + TB - 1) / TB, TB, 0, stream>>>(Hbuf, score, keys,
                                                               poolOut, newidOut, kkeep);
  };

  // ---- level 1 ----
  run_level(x, W1, b1, attn1, src1, dst1, /*mask*/nullptr, N, K1,
            /*H*/bufA, /*agg*/bufB, /*poolOut(x2)*/bufB, /*newid*/newid);
  edge_remap<<<(E + TB - 1) / TB, TB, 0, stream>>>(src1, dst1, nullptr, newid,
                                                   src2, dst2, mask2, E);
  // ---- level 2 ---- (x2 in bufB is consumed by the GEMM before bufB is reused as agg)
  run_level(bufB, W2, b2, attn2, src2, dst2, mask2, K1, K2,
            /*H*/bufA, /*agg*/bufB, /*poolOut*/(float*)d_out, /*newid*/nullptr);
}